// CausalAttentionPixelBlock_51393578664175
// MI455X (gfx1250) — compile-verified
//
#include <hip/hip_runtime.h>

// CDNA5 (gfx1250) flash-attention for tiny head dim using V_WMMA_F32_16X16X4_F32.
// wave32, one wave per 16-query tile, grid = (seq/16, heads, stacks).

typedef __attribute__((ext_vector_type(2))) float v2f;
typedef __attribute__((ext_vector_type(8))) float v8f;

#define SEQ   2048   // D*H*W = 8*16*16
#define CTOT  64     // channels
#define NH    8      // heads
#define CK    8      // channels per head
#define NSTK  3

__global__ __launch_bounds__(32)
void causal_attn_fa_f32wmma(const float* __restrict__ Kp,
                            const float* __restrict__ Qp,
                            const float* __restrict__ Vp,
                            float* __restrict__ Op) {
    const int lane = threadIdx.x;     // 0..31
    const int l16  = lane & 15;
    const int hi   = lane >> 4;       // which 16-lane half
    const int it   = blockIdx.x;      // query tile index (0..127)
    const int head = blockIdx.y;
    const int stk  = blockIdx.z;

    // channel-0 row of this (stack, head); B == 1
    const size_t base = ((size_t)stk * CTOT + (size_t)head * CK) * SEQ;
    const float* __restrict__ q = Qp + base;
    const float* __restrict__ k = Kp + base;
    const float* __restrict__ v = Vp + base;
    float*       __restrict__ o = Op + base;

    const int   ibase = it * 16;
    const int   iq    = ibase + l16;             // this lane's query position
    const float scale = 0.35355339059327373f;    // ck^-0.5 = 1/sqrt(8)

    // ---- Q in B-fragment layout (K'=4 channels x N=16 queries), two frags for ck=8.
    // B layout (f32, 4x16): vgpr0 = rows {0 | 2}, vgpr1 = rows {1 | 3} per half.
    const int c0 = hi * 2;
    v2f qb0, qb1;
    qb0.x = q[(size_t)(c0    ) * SEQ + iq] * scale;
    qb0.y = q[(size_t)(c0 + 1) * SEQ + iq] * scale;
    qb1.x = q[(size_t)(c0 + 4) * SEQ + iq] * scale;
    qb1.y = q[(size_t)(c0 + 5) * SEQ + iq] * scale;

    float m_run = -1e30f;       // running max (per query, replicated in both halves)
    float l_run = 0.0f;         // running denom
    v8f   o_acc = {0.f,0.f,0.f,0.f,0.f,0.f,0.f,0.f};   // O^T: channels x queries

    const int  cv    = l16;     // V A-fragment row (channel), rows 8..15 are zero pad
    const bool cv_ok = (cv < CK);

    for (int jt = 0; jt <= it; ++jt) {
        const int jbase = jt * 16;

        // ---- S^T = K_tile(16 keys x 8ch) x Q(8ch x 16 queries): 2 chained K=4 WMMAs.
        // A layout (f32, 16x4): lane holds M = lane%16; vgpr0 = K {0|2}, vgpr1 = K {1|3}.
        const int jk = jbase + l16;              // this lane's key row
        v2f ka0, ka1;
        ka0.x = k[(size_t)(c0    ) * SEQ + jk];
        ka0.y = k[(size_t)(c0 + 1) * SEQ + jk];
        ka1.x = k[(size_t)(c0 + 4) * SEQ + jk];
        ka1.y = k[(size_t)(c0 + 5) * SEQ + jk];

        v8f sacc = {0.f,0.f,0.f,0.f,0.f,0.f,0.f,0.f};
        sacc = __builtin_amdgcn_wmma_f32_16x16x4_f32(false, ka0, false, qb0,
                                                     (short)0, sacc, false, false);
        sacc = __builtin_amdgcn_wmma_f32_16x16x4_f32(false, ka1, false, qb1,
                                                     (short)0, sacc, false, false);

        // D layout: lane holds query col N = lane%16; key rows r+8*hi across 8 vgprs.
        float p[8];
        #pragma unroll
        for (int r = 0; r < 8; ++r) p[r] = sacc[r];

        if (jt == it) {                          // causal mask, diagonal tile only
            #pragma unroll
            for (int r = 0; r < 8; ++r) {
                const int key_abs = jbase + r + 8 * hi;
                if (key_abs > iq) p[r] = -1e30f;
            }
        }

        // ---- online softmax: per-lane max/sum over 8 keys, merge halves via xor-16.
        float mloc = p[0];
        #pragma unroll
        for (int r = 1; r < 8; ++r) mloc = fmaxf(mloc, p[r]);
        const float mtile = fmaxf(mloc, __shfl_xor(mloc, 16, 32));
        const float m_new = fmaxf(m_run, mtile);
        const float alpha = __expf(m_run - m_new);
        float lsum = 0.0f;
        #pragma unroll
        for (int r = 0; r < 8; ++r) { p[r] = __expf(p[r] - m_new); lsum += p[r]; }
        l_run = l_run * alpha + lsum + __shfl_xor(lsum, 16, 32);
        m_run = m_new;
        #pragma unroll
        for (int r = 0; r < 8; ++r) o_acc[r] *= alpha;   // rescale accumulator

        // ---- O^T += V_tile(chan x keys) x P^T(keys x queries): 4 chained K=4 WMMAs.
        #pragma unroll
        for (int t = 0; t < 4; ++t) {
            // Rebuild B fragment rows 4t..4t+3 from D-layout P via cross-half shuffles.
            v2f pb;
            if (t < 2) {                         // rows 0..7 live in half 0
                const float cx = __shfl_xor(p[4*t + 2], 16, 32);
                const float cy = __shfl_xor(p[4*t + 3], 16, 32);
                pb.x = hi ? cx : p[4*t    ];
                pb.y = hi ? cy : p[4*t + 1];
            } else {                             // rows 8..15 live in half 1
                const float cx = __shfl_xor(p[4*t - 8], 16, 32);
                const float cy = __shfl_xor(p[4*t - 7], 16, 32);
                pb.x = hi ? p[4*t - 6] : cx;
                pb.y = hi ? p[4*t - 5] : cy;
            }
            // A fragment: V channel row cv, keys {4t,4t+1 | 4t+2,4t+3} (contiguous b64).
            v2f va;
            if (cv_ok) {
                const float* vp = v + (size_t)cv * SEQ + jbase + 4*t + 2*hi;
                va.x = vp[0];
                va.y = vp[1];
            } else {
                va.x = 0.0f; va.y = 0.0f;
            }
            o_acc = __builtin_amdgcn_wmma_f32_16x16x4_f32(false, va, false, pb,
                                                          (short)0, o_acc, false, false);
        }
    }

    // ---- epilogue: normalize and store. Lanes 0..15 hold real channels 0..7.
    const float inv = 1.0f / l_run;
    if (hi == 0) {
        #pragma unroll
        for (int r = 0; r < 8; ++r) {
            o[(size_t)r * SEQ + iq] = o_acc[r] * inv;
        }
    }
}

extern "C" void kernel_launch(void* const* d_in, const int* in_sizes, int n_in,
                              void* d_out, int out_size, void* d_ws, size_t ws_size,
                              hipStream_t stream) {
    (void)in_sizes; (void)n_in; (void)d_ws; (void)ws_size; (void)out_size;
    // setup_inputs order: keys, queries, values, attn_mask (unused; causal computed
    // analytically), num_heads (fixed at 8 by the reference).
    const float* Kp = (const float*)d_in[0];
    const float* Qp = (const float*)d_in[1];
    const float* Vp = (const float*)d_in[2];
    float*       Op = (float*)d_out;

    dim3 grid(SEQ / 16, NH, NSTK);   // 128 x 8 x 3 = 3072 waves
    dim3 block(32);
    causal_attn_fa_f32wmma<<<grid, block, 0, stream>>>(Kp, Qp, Vp, Op);
}